// ShapeNetPretrainingLoss_11931419148399
// MI455X (gfx1250) — compile-verified
//
#include <hip/hip_runtime.h>
#include <hip/hip_bf16.h>
#include <cstdint>

typedef __attribute__((ext_vector_type(2))) float v2f;
typedef __attribute__((ext_vector_type(8))) float v8f;

#define BB 8
#define NN 4096
#define MM 4096
#define ROWS_PER_BLOCK 128   // 8 waves x 16 rows
#define THREADS 256

// ---------------------------------------------------------------------------
// Kernel 0: initialize global per-column min buffer to +inf (uint-encoded).
// d2 >= 0 after clamping, so IEEE-754 bit patterns are monotonic and integer
// atomicMin implements float min.
// ---------------------------------------------------------------------------
__global__ void init_colmin_kernel(unsigned int* __restrict__ colMin, int n) {
    int i = blockIdx.x * blockDim.x + threadIdx.x;
    if (i < n) colMin[i] = 0x7F800000u;  // +inf
}

// ---------------------------------------------------------------------------
// Kernel 1: tiled pairwise distances via V_WMMA_F32_16X16X4_F32.
//   A (16x4)  = [px, py, pz, |p|^2]   per row
//   B (4x16)  = [-2tx; -2ty; -2tz; 1] per column
//   D = A*B   = |p|^2 - 2 p.t  ;  d2 = D + |t|^2
// Each wave owns 16 rows; sweeps all 256 column tiles of 16.
// ---------------------------------------------------------------------------
__global__ __launch_bounds__(THREADS)
void chamfer_wmma_kernel(const float* __restrict__ pred,
                         const float* __restrict__ tgt,
                         float* __restrict__ rowMin,
                         unsigned int* __restrict__ colMin) {
    __shared__ unsigned int sCol[MM];  // 16 KB per-block column mins

    const int tid  = threadIdx.x;
    const int lane = tid & 31;
    const int wave = tid >> 5;
    const int b        = blockIdx.x / (NN / ROWS_PER_BLOCK);
    const int rowBlock = blockIdx.x % (NN / ROWS_PER_BLOCK);

    for (int i = tid; i < MM; i += THREADS) sCol[i] = 0x7F800000u;
    __syncthreads();

    // --- A operand: loaded once per wave ---------------------------------
    // A layout (32-bit 16x4): lanes 0-15 hold K=0,1 of M=lane; lanes 16-31
    // hold K=2,3 of M=lane-16.
    const int rowBase = rowBlock * ROWS_PER_BLOCK + wave * 16;
    const int row     = rowBase + (lane & 15);
    const float* pp = pred + ((size_t)b * NN + row) * 3;
    const float px = pp[0], py = pp[1], pz = pp[2];
    const float psq = px * px + py * py + pz * pz;
    v2f a;
    a.x = (lane < 16) ? px : pz;
    a.y = (lane < 16) ? py : psq;

    float rowAcc[8];
#pragma unroll
    for (int r = 0; r < 8; ++r) rowAcc[r] = __builtin_inff();

    for (int colBase = 0; colBase < MM; colBase += 16) {
        const int col = colBase + (lane & 15);
        const float* tp = tgt + ((size_t)b * MM + col) * 3;
        const float tx = tp[0], ty = tp[1], tz = tp[2];
        const float tsq = tx * tx + ty * ty + tz * tz;

        // B layout (4x16): lanes 0-15 hold K=0,1 rows at N=lane; lanes
        // 16-31 hold K=2,3 rows at N=lane-16 (mirrors the A layout).
        v2f bm;
        bm.x = (lane < 16) ? (-2.0f * tx) : (-2.0f * tz);
        bm.y = (lane < 16) ? (-2.0f * ty) : 1.0f;

        v8f c = {0.f, 0.f, 0.f, 0.f, 0.f, 0.f, 0.f, 0.f};
        // 8 args: (neg_a, A, neg_b, B, c_mod, C, reuse_a, reuse_b)
        c = __builtin_amdgcn_wmma_f32_16x16x4_f32(
                false, a, false, bm, (short)0, c, false, false);

        // C/D layout: VGPR r -> row r (lanes 0-15) / row 8+r (lanes 16-31),
        // column = lane & 15.
        float colv = __builtin_inff();
#pragma unroll
        for (int r = 0; r < 8; ++r) {
            float d2 = fmaxf(c[r] + tsq, 0.0f);
            rowAcc[r] = fminf(rowAcc[r], d2);
            colv = fminf(colv, d2);
        }
        // Two lanes (L, L+16) hit the same column; LDS atomic handles it.
        atomicMin(&sCol[col], __float_as_uint(colv));
    }

    // Row mins: reduce across the 16 lanes of each half.
#pragma unroll
    for (int r = 0; r < 8; ++r) {
        float v = rowAcc[r];
#pragma unroll
        for (int off = 1; off <= 8; off <<= 1)
            v = fminf(v, __shfl_xor(v, off, 32));
        if ((lane & 15) == 0) {
            const int rr = rowBase + r + ((lane >> 4) << 3);
            rowMin[(size_t)b * NN + rr] = v;
        }
    }

    __syncthreads();
    for (int m = tid; m < MM; m += THREADS)
        atomicMin(&colMin[(size_t)b * MM + m], sCol[m]);
}

// ---------------------------------------------------------------------------
// Kernel 2: final reductions -> scalar loss.
// ---------------------------------------------------------------------------
__global__ __launch_bounds__(1024)
void finalize_kernel(const float* __restrict__ rowMin,
                     const unsigned int* __restrict__ colMin,
                     const float* __restrict__ unc,
                     float* __restrict__ out) {
    const int tid = threadIdx.x;
    float sRow = 0.f, sEmd = 0.f, sColS = 0.f, sUnc = 0.f;

    for (int i = tid; i < BB * NN; i += 1024) {
        const float v = rowMin[i];
        sRow += v;
        sEmd += sqrtf(fmaxf(v, 0.f));
        sUnc += unc[i];
    }
    for (int i = tid; i < BB * MM; i += 1024)
        sColS += __uint_as_float(colMin[i]);

    __shared__ float red[4][32];
    float vals[4] = {sRow, sEmd, sColS, sUnc};
    const int lane = tid & 31, w = tid >> 5;
#pragma unroll
    for (int k = 0; k < 4; ++k) {
        float v = vals[k];
        for (int off = 16; off >= 1; off >>= 1) v += __shfl_down(v, off, 32);
        if (lane == 0) red[k][w] = v;
    }
    __syncthreads();
    if (tid < 32) {
#pragma unroll
        for (int k = 0; k < 4; ++k) {
            float v = red[k][tid];
            for (int off = 16; off >= 1; off >>= 1) v += __shfl_down(v, off, 32);
            vals[k] = v;
        }
        if (tid == 0) {
            const float invBN = 1.0f / (float)(BB * NN);
            const float invBM = 1.0f / (float)(BB * MM);
            const float chamfer = vals[0] * invBN + vals[2] * invBM;
            const float emd     = vals[1] * invBN;
            const float uncm    = vals[3] * invBN;
            out[0] = chamfer * 1.0f + emd * 0.5f + uncm * 0.01f;
        }
    }
}

// ---------------------------------------------------------------------------
extern "C" void kernel_launch(void* const* d_in, const int* in_sizes, int n_in,
                              void* d_out, int out_size, void* d_ws, size_t ws_size,
                              hipStream_t stream) {
    const float* pred = (const float*)d_in[0];       // [B,N,3]
    const float* tgt  = (const float*)d_in[1];       // [B,M,3]
    const float* unc  = (const float*)d_in[2];       // [B,N]

    float*        rowMin = (float*)d_ws;                                    // B*N floats
    unsigned int* colMin = (unsigned int*)((char*)d_ws +
                                           sizeof(float) * (size_t)BB * NN); // B*M uints

    init_colmin_kernel<<<(BB * MM + 255) / 256, 256, 0, stream>>>(colMin, BB * MM);
    chamfer_wmma_kernel<<<BB * (NN / ROWS_PER_BLOCK), THREADS, 0, stream>>>(
        pred, tgt, rowMin, colMin);
    finalize_kernel<<<1, 1024, 0, stream>>>(rowMin, colMin, unc, (float*)d_out);
}